// SinkhornKnopp_69698729279526
// MI455X (gfx1250) — compile-verified
//
#include <hip/hip_runtime.h>
#include <math.h>

typedef float v2f __attribute__((ext_vector_type(2)));
typedef float v8f __attribute__((ext_vector_type(8)));

#define SK_N     512
#define SK_BATCH 256
#define SK_ITERS 20
#define SK_EPS   1e-8f

// ---------------- Kernel 1: per-block partial max ----------------
__global__ __launch_bounds__(256) void k_partial_max(const float* __restrict__ m,
                                                     float* __restrict__ partial,
                                                     long total) {
  __shared__ float red[256];
  float mx = -INFINITY;
  long stride = (long)gridDim.x * (long)blockDim.x;
  for (long i = (long)blockIdx.x * blockDim.x + threadIdx.x; i < total; i += stride)
    mx = fmaxf(mx, m[i]);
  red[threadIdx.x] = mx;
  __syncthreads();
  for (int s = 128; s > 0; s >>= 1) {
    if (threadIdx.x < s) red[threadIdx.x] = fmaxf(red[threadIdx.x], red[threadIdx.x + s]);
    __syncthreads();
  }
  if (threadIdx.x == 0) partial[blockIdx.x] = red[0];
}

// ---------------- Kernel 2: final max (one block) ----------------
__global__ __launch_bounds__(256) void k_final_max(const float* __restrict__ partial,
                                                   float* __restrict__ gmax, int n) {
  __shared__ float red[256];
  float mx = -INFINITY;
  for (int i = threadIdx.x; i < n; i += 256) mx = fmaxf(mx, partial[i]);
  red[threadIdx.x] = mx;
  __syncthreads();
  for (int s = 128; s > 0; s >>= 1) {
    if (threadIdx.x < s) red[threadIdx.x] = fmaxf(red[threadIdx.x], red[threadIdx.x + s]);
    __syncthreads();
  }
  if (threadIdx.x == 0) gmax[0] = red[0];
}

// ---------------- Kernel 3: P0 = exp(M - gmax) -> out ----------------
__global__ __launch_bounds__(256) void k_exp(const float* __restrict__ m,
                                             const float* __restrict__ gmax,
                                             float* __restrict__ p, long total) {
  const float g = gmax[0];
  long stride = (long)gridDim.x * (long)blockDim.x;
  for (long i = (long)blockIdx.x * blockDim.x + threadIdx.x; i < total; i += stride)
    p[i] = expf(m[i] - g);
}

// ---------------- Kernel 4: per-matrix Sinkhorn via diag(u) P0 diag(v) ----------------
// One workgroup (256 threads = 8 waves) per matrix. 20 iterations of two WMMA
// matvec passes over the L2-resident matrix, then in-place final scaling.
__global__ __launch_bounds__(256) void k_sinkhorn(float* __restrict__ P) {
  __shared__ float u[SK_N], v[SK_N], rs[SK_N], cs[SK_N];

  const int tid  = threadIdx.x;
  const int lane = tid & 31;
  const int wave = tid >> 5;                 // 0..7
  float* __restrict__ Pm = P + (size_t)blockIdx.x * SK_N * SK_N;

  for (int i = tid; i < SK_N; i += 256) { u[i] = 1.0f; v[i] = 1.0f; }
  __syncthreads();

  // A-matrix 16x4 f32 layout: lanes 0-15 -> M=lane, VGPR0/1 = K0/K1;
  //                           lanes16-31 -> M=lane-16, VGPR0/1 = K2/K3.
  const int khalfA = (lane >> 4) * 2;        // 0 or 2
  // B-matrix 4x16 f32 layout: VGPR0 = rows K0 (lanes 0-15), K1 (lanes 16-31);
  //                           VGPR1 = rows K2, K3.
  const int khalfB = (lane >> 4);            // 0 or 1
  const int nA     = lane & 15;

  for (int it = 0; it < SK_ITERS; ++it) {
    // ---- Phase A: rs = P0 * v  (A = P0 row-tile, B = v broadcast to all 16 cols) ----
    for (int tile = wave; tile < SK_N / 16; tile += 8) {
      const int m0 = tile * 16;
      const float* __restrict__ rowp = Pm + (size_t)(m0 + nA) * SK_N;
      v8f acc = {};
      for (int k = 0; k < SK_N; k += 4) {
        v2f a, b;
        a.x = rowp[k + khalfA];
        a.y = rowp[k + khalfA + 1];
        b.x = v[k + khalfB];
        b.y = v[k + khalfB + 2];
        acc = __builtin_amdgcn_wmma_f32_16x16x4_f32(false, a, false, b,
                                                    (short)0, acc, false, false);
      }
      // D[M][N] identical across N; lane 0 holds rows m0..m0+7 (N=0),
      // lane 16 holds rows m0+8..m0+15.
      if (nA == 0) {
        const int mbase = m0 + (lane >> 4) * 8;
        rs[mbase + 0] = acc[0]; rs[mbase + 1] = acc[1];
        rs[mbase + 2] = acc[2]; rs[mbase + 3] = acc[3];
        rs[mbase + 4] = acc[4]; rs[mbase + 5] = acc[5];
        rs[mbase + 6] = acc[6]; rs[mbase + 7] = acc[7];
      }
    }
    __syncthreads();
    for (int i = tid; i < SK_N; i += 256) {
      const float ui = u[i];
      u[i] = ui / (ui * rs[i] + SK_EPS);
    }
    __syncthreads();

    // ---- Phase B: cs = P0^T * u  (A = u broadcast to all 16 rows, B = P0 tile) ----
    for (int tile = wave; tile < SK_N / 16; tile += 8) {
      const int j0 = tile * 16;
      v8f acc = {};
      for (int k = 0; k < SK_N; k += 4) {
        v2f a, b;
        a.x = u[k + khalfA];
        a.y = u[k + khalfA + 1];
        b.x = Pm[(size_t)(k + khalfB) * SK_N + j0 + nA];
        b.y = Pm[(size_t)(k + khalfB + 2) * SK_N + j0 + nA];
        acc = __builtin_amdgcn_wmma_f32_16x16x4_f32(false, a, false, b,
                                                    (short)0, acc, false, false);
      }
      // All rows of D identical; VGPR0 on lanes 0-15 is D[0][N=lane].
      if (lane < 16) cs[j0 + lane] = acc[0];
    }
    __syncthreads();
    for (int j = tid; j < SK_N; j += 256) {
      const float vj = v[j];
      v[j] = vj / (vj * cs[j] + SK_EPS);
    }
    __syncthreads();
  }

  // ---- Final: P = diag(u) * P0 * diag(v), in place ----
  for (int idx = tid; idx < SK_N * SK_N; idx += 256) {
    const int i = idx >> 9;       // / 512
    const int j = idx & (SK_N - 1);
    Pm[idx] = u[i] * Pm[idx] * v[j];
  }
}

extern "C" void kernel_launch(void* const* d_in, const int* in_sizes, int n_in,
                              void* d_out, int out_size, void* d_ws, size_t ws_size,
                              hipStream_t stream) {
  const float* M  = (const float*)d_in[0];
  float* out      = (float*)d_out;
  float* W        = (float*)d_ws;
  float* gmax     = W;          // 1 float
  float* partial  = W + 64;     // 1024 floats, 256B-aligned region

  const long total = (long)in_sizes[0];   // 256*512*512
  const int  nblk  = 1024;

  k_partial_max<<<nblk, 256, 0, stream>>>(M, partial, total);
  k_final_max  <<<1,    256, 0, stream>>>(partial, gmax, nblk);
  k_exp        <<<4096, 256, 0, stream>>>(M, gmax, out, total);
  k_sinkhorn   <<<SK_BATCH, 256, 0, stream>>>(out);
}